// GATGCNSelector_60275571032662
// MI455X (gfx1250) — compile-verified
//
#include <hip/hip_runtime.h>
#include <float.h>

#define IN_C 770
#define HID 128
#define NEG_SLOPE 0.2f
#define LN_EPS 1e-5f

typedef __attribute__((ext_vector_type(2))) float v2f;
typedef __attribute__((ext_vector_type(8))) float v8f;

// ---------------------------------------------------------------------------
// GEMM: xp = x @ W1   (M = nN, K = 770, N = 128) via V_WMMA_F32_16X16X4_F32
// ---------------------------------------------------------------------------
constexpr int KCHUNK = 64;
constexpr int ROWS_PER_BLOCK = 64;
constexpr int A_STRIDE = KCHUNK + 4;   // 68: avoids LDS bank conflicts on A reads
constexpr int B_STRIDE = HID + 4;      // 132

__global__ __launch_bounds__(256, 2)
void gat_gemm_xp(const float* __restrict__ x, const float* __restrict__ W1,
                 float* __restrict__ xp, int nN) {
  __shared__ float As[ROWS_PER_BLOCK * A_STRIDE];
  __shared__ float Bs[KCHUNK * B_STRIDE];
  const int tid  = threadIdx.x;
  const int lane = tid & 31;
  const int wave = tid >> 5;                 // 0..7 -> 16-col tile
  const int hl   = lane & 15;
  const int koff = (lane < 16) ? 0 : 2;      // f32 A/B frag K split across half-waves
  const int rowBase = blockIdx.x * ROWS_PER_BLOCK;
  const int colBase = wave * 16;

  v8f acc[4] = {};

  const int nChunks = (IN_C + KCHUNK - 1) / KCHUNK;   // 13 (zero-padded tail)
  for (int ch = 0; ch < nChunks; ++ch) {
    const int kBase = ch * KCHUNK;
    // cooperative load: A tile 64x64
    for (int i = tid; i < ROWS_PER_BLOCK * KCHUNK; i += 256) {
      int r = i >> 6, c = i & 63;
      int gr = rowBase + r, gk = kBase + c;
      As[r * A_STRIDE + c] =
          (gr < nN && gk < IN_C) ? x[(size_t)gr * IN_C + gk] : 0.0f;
    }
    // cooperative load: B tile 64x128
    for (int i = tid; i < KCHUNK * HID; i += 256) {
      int r = i >> 7, c = i & 127;
      int gk = kBase + r;
      Bs[r * B_STRIDE + c] = (gk < IN_C) ? W1[(size_t)gk * HID + c] : 0.0f;
    }
    __syncthreads();
    for (int kk = 0; kk < KCHUNK; kk += 4) {
      const int kr = kk + koff;
      // B 4x16 frag: lanes 0-15 rows K=kk+0/1, lanes 16-31 rows K=kk+2/3
      v2f bfrag = { Bs[kr * B_STRIDE + colBase + hl],
                    Bs[(kr + 1) * B_STRIDE + colBase + hl] };
#pragma unroll
      for (int t = 0; t < 4; ++t) {
        const int m = t * 16 + hl;
        // A 16x4 frag: lane holds row m, K=kr..kr+1
        v2f afrag = { As[m * A_STRIDE + kr], As[m * A_STRIDE + kr + 1] };
        acc[t] = __builtin_amdgcn_wmma_f32_16x16x4_f32(
            false, afrag, false, bfrag, (short)0, acc[t], false, false);
      }
    }
    __syncthreads();
  }
  // D 16x16 f32 layout: VGPR j -> rows j (lanes 0-15) / j+8 (lanes 16-31)
#pragma unroll
  for (int t = 0; t < 4; ++t) {
#pragma unroll
    for (int j = 0; j < 8; ++j) {
      int row = rowBase + t * 16 + ((lane < 16) ? j : (j + 8));
      int col = colBase + hl;
      if (row < nN) xp[(size_t)row * HID + col] = acc[t][j];
    }
  }
}

// ---------------------------------------------------------------------------
__device__ __forceinline__ float wave_sum(float v) {
#pragma unroll
  for (int off = 16; off > 0; off >>= 1) v += __shfl_xor(v, off, 32);
  return v;
}

__device__ __forceinline__ void get_edge(const long long* __restrict__ ei,
                                         long long E, long long idx,
                                         int& s, int& d) {
  if (idx < E) { s = (int)ei[idx]; d = (int)ei[E + idx]; }
  else         { s = d = (int)(idx - E); }        // self loops appended
}

// al_s = xp @ a_src, al_d = xp @ a_dst (one wave per node)
__global__ __launch_bounds__(256)
void node_alpha(const float* __restrict__ xp, const float* __restrict__ a_src,
                const float* __restrict__ a_dst, float* __restrict__ al_s,
                float* __restrict__ al_d, int nN) {
  int lane = threadIdx.x & 31;
  int node = blockIdx.x * 8 + (threadIdx.x >> 5);
  if (node >= nN) return;
  const float4 v  = ((const float4*)(xp + (size_t)node * HID))[lane];
  const float4 as = ((const float4*)a_src)[lane];
  const float4 ad = ((const float4*)a_dst)[lane];
  float ds = wave_sum(v.x * as.x + v.y * as.y + v.z * as.z + v.w * as.w);
  float dd = wave_sum(v.x * ad.x + v.y * ad.y + v.z * ad.z + v.w * ad.w);
  if (lane == 0) { al_s[node] = ds; al_d[node] = dd; }
}

__global__ __launch_bounds__(256)
void init_nodes(float* __restrict__ m, float* __restrict__ den,
                float* __restrict__ deg, float* __restrict__ out,
                const float* __restrict__ b2, int nN) {
  int i = blockIdx.x * 256 + threadIdx.x;
  if (i >= nN) return;
  m[i] = -FLT_MAX; den[i] = 0.0f; deg[i] = 0.0f; out[i] = b2[0];
}

// pass 1: leaky-relu edge logits, segment max, degree count
__global__ __launch_bounds__(256)
void edge_pre(const long long* __restrict__ ei, long long E, long long EN,
              const float* __restrict__ al_s, const float* __restrict__ al_d,
              float* __restrict__ ework, float* __restrict__ m,
              float* __restrict__ deg) {
  long long idx = (long long)blockIdx.x * 256 + threadIdx.x;
  if (idx >= EN) return;
  int s, d; get_edge(ei, E, idx, s, d);
  float e = al_s[s] + al_d[d];
  e = (e > 0.0f) ? e : NEG_SLOPE * e;
  ework[idx] = e;
  atomicMax(&m[d], e);          // global_atomic_max_num_f32
  atomicAdd(&deg[d], 1.0f);
}

// pass 2: num = exp(e - m[dst]), segment sum
__global__ __launch_bounds__(256)
void edge_exp(const long long* __restrict__ ei, long long E, long long EN,
              const float* __restrict__ m, float* __restrict__ ework,
              float* __restrict__ den) {
  long long idx = (long long)blockIdx.x * 256 + threadIdx.x;
  if (idx >= EN) return;
  int s, d; get_edge(ei, E, idx, s, d);
  float num = expf(ework[idx] - m[d]);
  ework[idx] = num;
  atomicAdd(&den[d], num);
}

// pass 3: agg[dst] += alpha * xp[src]   (one wave per edge, 4 ch/lane)
__global__ __launch_bounds__(256)
void edge_agg(const long long* __restrict__ ei, long long E, long long EN,
              const float* __restrict__ ework, const float* __restrict__ den,
              const float* __restrict__ xp, float* __restrict__ agg) {
  int lane = threadIdx.x & 31;
  long long idx = (long long)blockIdx.x * 8 + (threadIdx.x >> 5);
  if (idx >= EN) return;
  int s, d; get_edge(ei, E, idx, s, d);
  float alpha = ework[idx] / den[d];
  float4 v = ((const float4*)(xp + (size_t)s * HID))[lane];
  float* ag = agg + (size_t)d * HID + lane * 4;
  atomicAdd(ag + 0, alpha * v.x);
  atomicAdd(ag + 1, alpha * v.y);
  atomicAdd(ag + 2, alpha * v.z);
  atomicAdd(ag + 3, alpha * v.w);
}

// fused: +b1, ReLU, LayerNorm, h@W2, deg->rsqrt   (one wave per node)
__global__ __launch_bounds__(256)
void node_post(const float* __restrict__ agg, const float* __restrict__ b1,
               const float* __restrict__ gamma, const float* __restrict__ beta,
               const float* __restrict__ W2, const float* __restrict__ deg,
               float* __restrict__ xp2, float* __restrict__ dinv, int nN) {
  int lane = threadIdx.x & 31;
  int node = blockIdx.x * 8 + (threadIdx.x >> 5);
  if (node >= nN) return;
  float4 v = ((const float4*)(agg + (size_t)node * HID))[lane];
  float4 bb = ((const float4*)b1)[lane];
  v.x = fmaxf(v.x + bb.x, 0.0f); v.y = fmaxf(v.y + bb.y, 0.0f);
  v.z = fmaxf(v.z + bb.z, 0.0f); v.w = fmaxf(v.w + bb.w, 0.0f);
  float s1 = wave_sum(v.x + v.y + v.z + v.w);
  float s2 = wave_sum(v.x * v.x + v.y * v.y + v.z * v.z + v.w * v.w);
  float mu = s1 * (1.0f / HID);
  float var = s2 * (1.0f / HID) - mu * mu;
  float rs = rsqrtf(var + LN_EPS);
  float4 g = ((const float4*)gamma)[lane];
  float4 be = ((const float4*)beta)[lane];
  float4 w = ((const float4*)W2)[lane];
  float hx = (v.x - mu) * rs * g.x + be.x;
  float hy = (v.y - mu) * rs * g.y + be.y;
  float hz = (v.z - mu) * rs * g.z + be.z;
  float hw = (v.w - mu) * rs * g.w + be.w;
  float dot = wave_sum(hx * w.x + hy * w.y + hz * w.z + hw * w.w);
  if (lane == 0) {
    xp2[node] = dot;
    float dg = deg[node];
    dinv[node] = (dg > 0.0f) ? rsqrtf(dg) : 0.0f;
  }
}

// GCN scatter: out[dst] += dinv[src]*dinv[dst]*xp2[src]
__global__ __launch_bounds__(256)
void edge_gcn(const long long* __restrict__ ei, long long E, long long EN,
              const float* __restrict__ dinv, const float* __restrict__ xp2,
              float* __restrict__ out) {
  long long idx = (long long)blockIdx.x * 256 + threadIdx.x;
  if (idx >= EN) return;
  int s, d; get_edge(ei, E, idx, s, d);
  atomicAdd(&out[d], dinv[s] * dinv[d] * xp2[s]);
}

// ---------------------------------------------------------------------------
extern "C" void kernel_launch(void* const* d_in, const int* in_sizes, int n_in,
                              void* d_out, int out_size, void* d_ws, size_t ws_size,
                              hipStream_t stream) {
  const float*     x     = (const float*)d_in[0];
  const long long* ei    = (const long long*)d_in[1];
  const float*     W1    = (const float*)d_in[2];
  const float*     a_src = (const float*)d_in[3];
  const float*     a_dst = (const float*)d_in[4];
  const float*     b1    = (const float*)d_in[5];
  const float*     gamma = (const float*)d_in[6];
  const float*     beta  = (const float*)d_in[7];
  const float*     W2    = (const float*)d_in[8];
  const float*     b2    = (const float*)d_in[9];
  float* out = (float*)d_out;

  const int nN = in_sizes[0] / IN_C;          // 100000
  const long long E = in_sizes[1] / 2;        // 1600000
  const long long EN = E + nN;                // edges + self loops

  // workspace carve-up (floats)
  float* ws = (float*)d_ws;
  size_t o = 0;
  float* xp    = ws + o; o += (size_t)nN * HID;
  float* agg   = ws + o; o += (size_t)nN * HID;
  float* al_s  = ws + o; o += nN;
  float* al_d  = ws + o; o += nN;
  float* segm  = ws + o; o += nN;
  float* den   = ws + o; o += nN;
  float* deg   = ws + o; o += nN;
  float* dinv  = ws + o; o += nN;
  float* xp2   = ws + o; o += nN;
  float* ework = ws + o; o += (size_t)EN;

  hipMemsetAsync(agg, 0, (size_t)nN * HID * sizeof(float), stream);

  int nodeBlocks = (nN + 255) / 256;
  init_nodes<<<nodeBlocks, 256, 0, stream>>>(segm, den, deg, out, b2, nN);

  int gemmBlocks = (nN + ROWS_PER_BLOCK - 1) / ROWS_PER_BLOCK;
  gat_gemm_xp<<<gemmBlocks, 256, 0, stream>>>(x, W1, xp, nN);

  int waveNodeBlocks = (nN + 7) / 8;
  node_alpha<<<waveNodeBlocks, 256, 0, stream>>>(xp, a_src, a_dst, al_s, al_d, nN);

  int edgeBlocks = (int)((EN + 255) / 256);
  edge_pre<<<edgeBlocks, 256, 0, stream>>>(ei, E, EN, al_s, al_d, ework, segm, deg);
  edge_exp<<<edgeBlocks, 256, 0, stream>>>(ei, E, EN, segm, ework, den);

  int edgeWaveBlocks = (int)((EN + 7) / 8);
  edge_agg<<<edgeWaveBlocks, 256, 0, stream>>>(ei, E, EN, ework, den, xp, agg);

  node_post<<<waveNodeBlocks, 256, 0, stream>>>(agg, b1, gamma, beta, W2, deg,
                                                xp2, dinv, nN);
  edge_gcn<<<edgeBlocks, 256, 0, stream>>>(ei, E, EN, dinv, xp2, out);
}